// GATNet_40535901339689
// MI455X (gfx1250) — compile-verified
//
#include <hip/hip_runtime.h>

typedef __attribute__((ext_vector_type(2))) float v2f;
typedef __attribute__((ext_vector_type(8))) float v8f;

#define N_NODES 20000
#define N_EDGES 320000
#define NEG_SLOPE 0.2f

// ---------------------------------------------------------------------------
// float atomic max via integer punning (works for mixed-sign values)
// ---------------------------------------------------------------------------
__device__ __forceinline__ void atomicMaxF(float* addr, float val) {
    if (val >= 0.0f) {
        atomicMax((int*)addr, __float_as_int(val));
    } else {
        atomicMin((unsigned int*)addr, __float_as_uint(val));
    }
}

// ---------------------------------------------------------------------------
// fill
// ---------------------------------------------------------------------------
__global__ void fill_kernel(float* __restrict__ p, float v, long long n) {
    long long i = (long long)blockIdx.x * blockDim.x + threadIdx.x;
    if (i < n) p[i] = v;
}

// ---------------------------------------------------------------------------
// D[M,N] = A[M,K] * B[K,N], row-major. M%16==0, N%16==0, K%8==0.
// One 16x16 output tile per wave via V_WMMA_F32_16X16X4_F32 (exact fp32).
// A layout: lane<16 -> row=lane, K=k+0/k+1 ; lane>=16 -> row=lane-16, K=k+2/k+3
// B layout (transposed mirror): lane<16 -> col=lane, rows k+0/k+1 ; etc.
// C/D layout: VGPR r -> row r (lanes 0-15) / row r+8 (lanes 16-31), col = lane%16
// Inner loop uses pointer stepping (no per-iter 64-bit index rebuild) and
// prefetches the contiguous A stream (global_prefetch_b8).
// ---------------------------------------------------------------------------
__global__ void gemm_wmma_f32(const float* __restrict__ A,
                              const float* __restrict__ B,
                              float* __restrict__ D,
                              int M, int N, int K)
{
    const int wave = threadIdx.x >> 5;        // 4 waves per block
    const int lane = threadIdx.x & 31;
    const int half = lane >> 4;               // 0 or 1
    const int l16  = lane & 15;

    const int m0 = blockIdx.y * 16;                   // M/16 tiles
    const int n0 = (blockIdx.x * 4 + wave) * 16;      // N/16 tiles, 4 per block

    const float* Ap = A + (size_t)(m0 + l16) * K + 2 * half;
    const float* Bp = B + (size_t)(2 * half) * N + (n0 + l16);
    const int bstep = 4 * N;                  // B advances 4 rows per k-step

    v8f acc = {0.f, 0.f, 0.f, 0.f, 0.f, 0.f, 0.f, 0.f};

    for (int k = 0; k < K; k += 8) {
        __builtin_prefetch(Ap + k + 64, 0, 0);   // next 256B of this A row
        v2f a0, a1, b0, b1;
        a0.x = Ap[k + 0];
        a0.y = Ap[k + 1];
        a1.x = Ap[k + 4];
        a1.y = Ap[k + 5];
        b0.x = Bp[0];
        b0.y = Bp[N];
        b1.x = Bp[bstep];
        b1.y = Bp[bstep + N];
        Bp += 2 * bstep;
        acc = __builtin_amdgcn_wmma_f32_16x16x4_f32(false, a0, false, b0,
                                                    (short)0, acc, false, false);
        acc = __builtin_amdgcn_wmma_f32_16x16x4_f32(false, a1, false, b1,
                                                    (short)0, acc, false, false);
    }

    float* Dp = D + (size_t)(m0 + 8 * half) * N + (n0 + l16);
#pragma unroll
    for (int r = 0; r < 8; ++r) {
        *Dp = acc[r];
        Dp += N;
    }
}

// ---------------------------------------------------------------------------
// es[n,h] = sum_c h[n,h,c]*a_src[h,c] ; ed likewise
// ---------------------------------------------------------------------------
__global__ void attn_coeff(const float* __restrict__ h,
                           const float* __restrict__ a_s,
                           const float* __restrict__ a_d,
                           float* __restrict__ es, float* __restrict__ ed,
                           int Nn, int H, int C)
{
    long long idx = (long long)blockIdx.x * blockDim.x + threadIdx.x;
    if (idx >= (long long)Nn * H) return;
    int n  = (int)(idx / H);
    int hh = (int)(idx % H);
    const float* hp  = h + (size_t)n * H * C + (size_t)hh * C;
    const float* asp = a_s + (size_t)hh * C;
    const float* adp = a_d + (size_t)hh * C;
    float s = 0.f, d = 0.f;
    for (int c = 0; c < C; ++c) {
        float v = hp[c];
        s += v * asp[c];
        d += v * adp[c];
    }
    es[idx] = s;
    ed[idx] = d;
}

// ---------------------------------------------------------------------------
// Edge pass 1: e = leaky_relu(es[src]+ed[dst]); segment max over dst
// Edges e < E come from edge lists; e in [E, ET) are self-loops (node e-E).
// ---------------------------------------------------------------------------
__global__ void edge_lrelu_max(const int* __restrict__ src, const int* __restrict__ dst,
                               int E, long long ET, int H,
                               const float* __restrict__ es, const float* __restrict__ ed,
                               float* __restrict__ ebuf, float* __restrict__ mx)
{
    long long idx = (long long)blockIdx.x * blockDim.x + threadIdx.x;
    if (idx >= ET * H) return;
    long long e = idx / H;
    int hh = (int)(idx % H);
    int s = (e < E) ? src[e] : (int)(e - E);
    int d = (e < E) ? dst[e] : (int)(e - E);
    float v = es[(size_t)s * H + hh] + ed[(size_t)d * H + hh];
    v = (v > 0.f) ? v : NEG_SLOPE * v;
    ebuf[idx] = v;
    atomicMaxF(&mx[(size_t)d * H + hh], v);
}

// ---------------------------------------------------------------------------
// Edge pass 2: ex = exp(e - max[dst]); segment sum over dst
// ---------------------------------------------------------------------------
__global__ void edge_expsum(const int* __restrict__ dst,
                            int E, long long ET, int H,
                            float* __restrict__ ebuf,
                            const float* __restrict__ mx,
                            float* __restrict__ den)
{
    long long idx = (long long)blockIdx.x * blockDim.x + threadIdx.x;
    if (idx >= ET * H) return;
    long long e = idx / H;
    int hh = (int)(idx % H);
    int d = (e < E) ? dst[e] : (int)(e - E);
    float ex = expf(ebuf[idx] - mx[(size_t)d * H + hh]);
    ebuf[idx] = ex;
    atomicAdd(&den[(size_t)d * H + hh], ex);
}

// ---------------------------------------------------------------------------
// Edge pass 3: out[dst,h,c] += (ex/den[dst,h]) * h[src,h,c]
// ---------------------------------------------------------------------------
__global__ void edge_aggregate(const int* __restrict__ src, const int* __restrict__ dst,
                               int E, long long ET, int H, int C,
                               const float* __restrict__ hbuf,
                               const float* __restrict__ ebuf,
                               const float* __restrict__ den,
                               float* __restrict__ out)
{
    const int HC = H * C;
    long long idx = (long long)blockIdx.x * blockDim.x + threadIdx.x;
    if (idx >= ET * HC) return;
    long long e = idx / HC;
    int f  = (int)(idx % HC);
    int hh = f / C;
    int s = (e < E) ? src[e] : (int)(e - E);
    int d = (e < E) ? dst[e] : (int)(e - E);
    float alpha = ebuf[e * H + hh] / den[(size_t)d * H + hh];
    atomicAdd(&out[(size_t)d * HC + f], alpha * hbuf[(size_t)s * HC + f]);
}

// ---------------------------------------------------------------------------
// out = elu(out + b)   (b broadcast along feature dim)
// ---------------------------------------------------------------------------
__global__ void bias_elu(float* __restrict__ out, const float* __restrict__ b,
                         long long n, int Fout)
{
    long long idx = (long long)blockIdx.x * blockDim.x + threadIdx.x;
    if (idx >= n) return;
    float v = out[idx] + b[idx % Fout];
    out[idx] = (v > 0.f) ? v : (expf(v) - 1.f);
}

// ---------------------------------------------------------------------------
// Layer-4 GEMV: h4[n] = dot(x[n,:], w)  (one wave per node, shuffle reduce)
// Also emits es4 = h4*a_s, ed4 = h4*a_d.
// ---------------------------------------------------------------------------
__global__ void gemv_l4(const float* __restrict__ x, const float* __restrict__ w,
                        const float* __restrict__ a_s, const float* __restrict__ a_d,
                        float* __restrict__ h, float* __restrict__ es,
                        float* __restrict__ ed, int Nn, int K)
{
    int node = blockIdx.x * (blockDim.x >> 5) + (threadIdx.x >> 5);
    int lane = threadIdx.x & 31;
    if (node >= Nn) return;
    const float* xp = x + (size_t)node * K;
    float s = 0.f;
    for (int k = lane; k < K; k += 32) s += xp[k] * w[k];
#pragma unroll
    for (int off = 16; off > 0; off >>= 1) s += __shfl_down(s, off);
    if (lane == 0) {
        h[node]  = s;
        es[node] = s * a_s[0];
        ed[node] = s * a_d[0];
    }
}

__global__ void final_sigmoid(const float* __restrict__ agg, const float* __restrict__ b4,
                              float* __restrict__ out, int Nn)
{
    int i = blockIdx.x * blockDim.x + threadIdx.x;
    if (i >= Nn) return;
    float v = agg[i] + b4[0];
    out[i] = 1.f / (1.f + expf(-v));
}

// ---------------------------------------------------------------------------
// Host orchestration
// ---------------------------------------------------------------------------
extern "C" void kernel_launch(void* const* d_in, const int* in_sizes, int n_in,
                              void* d_out, int out_size, void* d_ws, size_t ws_size,
                              hipStream_t stream)
{
    const float* x    = (const float*)d_in[0];
    const int*   esrc = (const int*)  d_in[1];
    const int*   edst = (const int*)  d_in[2];
    const float* W1   = (const float*)d_in[3];
    const float* a1s  = (const float*)d_in[4];
    const float* a1d  = (const float*)d_in[5];
    const float* b1   = (const float*)d_in[6];
    const float* W2   = (const float*)d_in[7];
    const float* a2s  = (const float*)d_in[8];
    const float* a2d  = (const float*)d_in[9];
    const float* b2   = (const float*)d_in[10];
    const float* W3   = (const float*)d_in[11];
    const float* a3s  = (const float*)d_in[12];
    const float* a3d  = (const float*)d_in[13];
    const float* b3   = (const float*)d_in[14];
    const float* W4   = (const float*)d_in[15];
    const float* a4s  = (const float*)d_in[16];
    const float* a4d  = (const float*)d_in[17];
    const float* b4   = (const float*)d_in[18];
    float* out = (float*)d_out;

    const int N = N_NODES, E = N_EDGES;
    const long long ET = (long long)E + N;   // + self loops

    // workspace layout (floats)
    float* bufA = (float*)d_ws;                      // 20000*1024
    float* bufB = bufA + (size_t)N * 1024;           // 20000*1024
    float* bufC = bufB + (size_t)N * 1024;           // 20000*512
    float* es   = bufC + (size_t)N * 512;            // 20000*8
    float* ed   = es   + (size_t)N * 8;
    float* mx   = ed   + (size_t)N * 8;
    float* den  = mx   + (size_t)N * 8;
    float* ebuf = den  + (size_t)N * 8;              // 340000*8
    float* agg  = ebuf + (size_t)ET * 8;             // 20000

    auto cdiv = [](long long a, long long b) { return (unsigned)((a + b - 1) / b); };

    auto run_layer = [&](const float* xin, const float* W, const float* as_,
                         const float* ad_, const float* bias,
                         int K, int H, int C, float* hbuf, float* obuf) {
        const int Fout = H * C;
        const long long NH = (long long)N * H;
        const long long NF = (long long)N * Fout;
        gemm_wmma_f32<<<dim3(Fout / 64, N / 16), 128, 0, stream>>>(xin, W, hbuf, N, Fout, K);
        attn_coeff<<<cdiv(NH, 256), 256, 0, stream>>>(hbuf, as_, ad_, es, ed, N, H, C);
        fill_kernel<<<cdiv(NH, 256), 256, 0, stream>>>(mx, -1e30f, NH);
        fill_kernel<<<cdiv(NH, 256), 256, 0, stream>>>(den, 0.f, NH);
        fill_kernel<<<cdiv(NF, 256), 256, 0, stream>>>(obuf, 0.f, NF);
        edge_lrelu_max<<<cdiv(ET * H, 256), 256, 0, stream>>>(esrc, edst, E, ET, H,
                                                              es, ed, ebuf, mx);
        edge_expsum<<<cdiv(ET * H, 256), 256, 0, stream>>>(edst, E, ET, H, ebuf, mx, den);
        edge_aggregate<<<cdiv(ET * H * C, 256), 256, 0, stream>>>(esrc, edst, E, ET, H, C,
                                                                  hbuf, ebuf, den, obuf);
        bias_elu<<<cdiv(NF, 256), 256, 0, stream>>>(obuf, bias, NF, Fout);
    };

    // conv1: 128 -> 8x32 ; conv2: 256 -> 8x64 ; conv3: 512 -> 8x128
    run_layer(x,    W1, a1s, a1d, b1, 128, 8, 32,  bufA, bufB);
    run_layer(bufB, W2, a2s, a2d, b2, 256, 8, 64,  bufA, bufC);
    run_layer(bufC, W3, a3s, a3d, b3, 512, 8, 128, bufA, bufB);

    // conv4: 1024 -> 1, heads=1, concat=False (mean over 1 head == identity)
    gemv_l4<<<cdiv(N, 8), 256, 0, stream>>>(bufB, W4, a4s, a4d, bufC, es, ed, N, 1024);
    fill_kernel<<<cdiv(N, 256), 256, 0, stream>>>(mx, -1e30f, N);
    fill_kernel<<<cdiv(N, 256), 256, 0, stream>>>(den, 0.f, N);
    fill_kernel<<<cdiv(N, 256), 256, 0, stream>>>(agg, 0.f, N);
    edge_lrelu_max<<<cdiv(ET, 256), 256, 0, stream>>>(esrc, edst, E, ET, 1, es, ed, ebuf, mx);
    edge_expsum<<<cdiv(ET, 256), 256, 0, stream>>>(edst, E, ET, 1, ebuf, mx, den);
    edge_aggregate<<<cdiv(ET, 256), 256, 0, stream>>>(esrc, edst, E, ET, 1, 1,
                                                      bufC, ebuf, den, agg);
    final_sigmoid<<<cdiv(N, 256), 256, 0, stream>>>(agg, b4, out, N);
}